// GATNeigh_Agg_44822278701142
// MI455X (gfx1250) — compile-verified
//
#include <hip/hip_runtime.h>
#include <hip/hip_bf16.h>
#include <cstddef>

// Problem constants (from reference setup_inputs)
constexpr int B = 4;
constexpr int E = 512;
constexpr int C = 64;
constexpr float SLOPE = 0.2f;
constexpr float NEG_INF = -1.0e12f;

typedef float v2f __attribute__((ext_vector_type(2)));
typedef float v8f __attribute__((ext_vector_type(8)));

__device__ __forceinline__ float leaky(float x) {
    return x > 0.f ? x : SLOPE * x;
}

// ---------------------------------------------------------------------------
// Kernel 1: vu = X @ U, iv = X @ W   (fp32 WMMA 16x16x4)
// One wave (32 threads) per (batch, 16-row tile). Computes a 16x64 slab of
// BOTH projections: 4 column tiles x 16 K-steps x 2 matrices = 128 WMMAs,
// 8 independent accumulators per K-step for co-execution.
// ---------------------------------------------------------------------------
__global__ __launch_bounds__(32) void proj_kernel(
    const float* __restrict__ x,   // [B,E,C]
    const float* __restrict__ W,   // [C,C]
    const float* __restrict__ U,   // [C,C]
    float* __restrict__ iv,        // [B,E,C]  (X@W)
    float* __restrict__ vu)        // [B,E,C]  (X@U)
{
    const int tile = blockIdx.x;          // B * (E/16) tiles
    const int b    = tile / (E / 16);
    const int rt   = tile % (E / 16);
    const int lane = threadIdx.x;

    const int rc    = lane & 15;          // row (A) / col (B) index
    const int khalf = (lane >> 4) * 2;    // K sub-offset: lanes 16-31 hold K=2,3

    const float* xrow = x + ((size_t)b * E + rt * 16 + rc) * C;

    v8f accW[4] = {v8f{}, v8f{}, v8f{}, v8f{}};
    v8f accU[4] = {v8f{}, v8f{}, v8f{}, v8f{}};

    for (int k0 = 0; k0 < C; k0 += 4) {
        // A tile 16x4: two K-contiguous floats per lane -> b64 load
        v2f a = *reinterpret_cast<const v2f*>(xrow + k0 + khalf);
#pragma unroll
        for (int nt = 0; nt < 4; ++nt) {
            const int col = nt * 16 + rc;
            v2f bw, bu;
            bw.x = W[(k0 + khalf) * C + col];
            bw.y = W[(k0 + khalf + 1) * C + col];
            bu.x = U[(k0 + khalf) * C + col];
            bu.y = U[(k0 + khalf + 1) * C + col];
            accW[nt] = __builtin_amdgcn_wmma_f32_16x16x4_f32(
                false, a, false, bw, (short)0, accW[nt], false, false);
            accU[nt] = __builtin_amdgcn_wmma_f32_16x16x4_f32(
                false, a, false, bu, (short)0, accU[nt], false, false);
        }
    }

    // D layout: VGPR r, lanes 0-15 -> (M=r, N=lane); lanes 16-31 -> (M=r+8)
    const int n     = lane & 15;
    const int mBase = (lane >> 4) * 8;
#pragma unroll
    for (int nt = 0; nt < 4; ++nt) {
#pragma unroll
        for (int r = 0; r < 8; ++r) {
            const size_t idx = ((size_t)b * E + rt * 16 + mBase + r) * C + nt * 16 + n;
            iv[idx] = accW[nt][r];
            vu[idx] = accU[nt][r];
        }
    }
}

// ---------------------------------------------------------------------------
// Kernel 2: logits + mask + softmax over j -> alpha[b,i,:]
// One block per (b,i); 256 threads, each owns j and j+256.
// logit = sum_c leaky(vu[i,c] + iv[j,c]) * yita[c]   (not GEMM-factorable:
// the LeakyReLU sits inside the c-reduction, so the |.| part stays pairwise)
// ---------------------------------------------------------------------------
__global__ __launch_bounds__(256) void attn_softmax_kernel(
    const float* __restrict__ vu,    // [B,E,C]
    const float* __restrict__ iv,    // [B,E,C]
    const int*   __restrict__ adj,   // [B,E,E]
    const float* __restrict__ yita,  // [C,1]
    float* __restrict__ alpha)       // [B,E,E]
{
    const int i = blockIdx.x;
    const int b = blockIdx.y;
    const int t = threadIdx.x;

    __shared__ float s_vu[C];
    __shared__ float s_yt[C];
    __shared__ float s_red[8];

    if (t < C) {
        s_vu[t] = vu[((size_t)b * E + i) * C + t];
        s_yt[t] = yita[t];
    }
    __syncthreads();

    float lg[2];
#pragma unroll
    for (int p = 0; p < 2; ++p) {
        const int j = t + p * 256;
        const float4* iv4 =
            reinterpret_cast<const float4*>(iv + ((size_t)b * E + j) * C);
        float acc = 0.f;
#pragma unroll
        for (int c4 = 0; c4 < C / 4; ++c4) {
            const float4 v = iv4[c4];
            const int c = c4 * 4;
            acc = fmaf(leaky(s_vu[c + 0] + v.x), s_yt[c + 0], acc);
            acc = fmaf(leaky(s_vu[c + 1] + v.y), s_yt[c + 1], acc);
            acc = fmaf(leaky(s_vu[c + 2] + v.z), s_yt[c + 2], acc);
            acc = fmaf(leaky(s_vu[c + 3] + v.w), s_yt[c + 3], acc);
        }
        const bool conn = adj[((size_t)b * E + i) * E + j] > 0;
        lg[p] = conn ? acc : NEG_INF;
    }

    const int wid = t >> 5;
    const int lid = t & 31;

    // --- block max ---
    float m = fmaxf(lg[0], lg[1]);
#pragma unroll
    for (int off = 16; off > 0; off >>= 1)
        m = fmaxf(m, __shfl_xor(m, off, 32));
    if (lid == 0) s_red[wid] = m;
    __syncthreads();
    if (t == 0) {
        float v = s_red[0];
#pragma unroll
        for (int w = 1; w < 8; ++w) v = fmaxf(v, s_red[w]);
        s_red[0] = v;
    }
    __syncthreads();
    const float rowmax = s_red[0];
    __syncthreads();

    // --- exp + block sum ---
    const float e0 = __expf(lg[0] - rowmax);
    const float e1 = __expf(lg[1] - rowmax);
    float s = e0 + e1;
#pragma unroll
    for (int off = 16; off > 0; off >>= 1)
        s += __shfl_xor(s, off, 32);
    if (lid == 0) s_red[wid] = s;
    __syncthreads();
    if (t == 0) {
        float v = s_red[0];
#pragma unroll
        for (int w = 1; w < 8; ++w) v += s_red[w];
        s_red[0] = v;
    }
    __syncthreads();
    const float inv = 1.f / s_red[0];

    float* arow = alpha + ((size_t)b * E + i) * E;
    arow[t]       = e0 * inv;
    arow[t + 256] = e1 * inv;
}

// ---------------------------------------------------------------------------
// Kernel 3: out = leaky(alpha @ X)   (fp32 WMMA 16x16x4, K=512 -> 128 steps)
// One wave per (batch, 16-row tile): computes the full 16x64 output slab with
// 4 independent accumulators per K-step (breaks the serial WMMA RAW chain,
// enables XDL co-execution) and shares each A load across all 4 column tiles.
// 128 waves x 512 WMMAs each.
// ---------------------------------------------------------------------------
__global__ __launch_bounds__(32) void agg_kernel(
    const float* __restrict__ alpha,  // [B,E,E]
    const float* __restrict__ x,      // [B,E,C]
    float* __restrict__ out)          // [B,E,C]
{
    const int id  = blockIdx.x;                 // B * (E/16)
    const int b   = id / (E / 16);
    const int it  = id % (E / 16);              // output row tile
    const int lane = threadIdx.x;

    const int rc    = lane & 15;
    const int khalf = (lane >> 4) * 2;

    const float* arow = alpha + ((size_t)b * E + it * 16 + rc) * E;
    const float* xb   = x + (size_t)b * E * C;

    v8f acc[4] = {v8f{}, v8f{}, v8f{}, v8f{}};

    for (int k0 = 0; k0 < E; k0 += 4) {
        // A tile 16x4: two K-contiguous floats per lane -> b64 load
        v2f a = *reinterpret_cast<const v2f*>(arow + k0 + khalf);
        const float* xk0 = xb + (size_t)(k0 + khalf) * C;
        const float* xk1 = xk0 + C;
#pragma unroll
        for (int nt = 0; nt < 4; ++nt) {
            const int col = nt * 16 + rc;
            v2f bb;
            bb.x = xk0[col];
            bb.y = xk1[col];
            acc[nt] = __builtin_amdgcn_wmma_f32_16x16x4_f32(
                false, a, false, bb, (short)0, acc[nt], false, false);
        }
    }

    const int n     = lane & 15;
    const int mBase = (lane >> 4) * 8;
#pragma unroll
    for (int nt = 0; nt < 4; ++nt) {
#pragma unroll
        for (int r = 0; r < 8; ++r) {
            const size_t idx =
                ((size_t)b * E + it * 16 + mBase + r) * C + nt * 16 + n;
            out[idx] = leaky(acc[nt][r]);
        }
    }
}

// ---------------------------------------------------------------------------
// Host launcher
// ---------------------------------------------------------------------------
extern "C" void kernel_launch(void* const* d_in, const int* in_sizes, int n_in,
                              void* d_out, int out_size, void* d_ws, size_t ws_size,
                              hipStream_t stream) {
    (void)in_sizes; (void)n_in; (void)out_size; (void)ws_size;

    const float* edge_attr = (const float*)d_in[0];  // [B,E,C]
    const int*   edge_adj  = (const int*)  d_in[1];  // [B,E,E]
    // d_in[2] = e_max (unused), d_in[3] = mask (unused)
    const float* W_2       = (const float*)d_in[4];  // [C,C]
    const float* U_2       = (const float*)d_in[5];  // [C,C]
    const float* yita      = (const float*)d_in[6];  // [C,1]
    float*       out       = (float*)d_out;          // [B,E,C]

    // Workspace carve-up
    char* ws = (char*)d_ws;
    float* vu    = (float*)(ws);                                         // B*E*C
    float* iv    = (float*)(ws + sizeof(float) * (size_t)B * E * C);     // B*E*C
    float* alpha = (float*)(ws + sizeof(float) * 2 * (size_t)B * E * C); // B*E*E

    // 1) projections (WMMA f32)
    proj_kernel<<<B * (E / 16), 32, 0, stream>>>(edge_attr, W_2, U_2, iv, vu);

    // 2) pairwise logits + mask + softmax
    attn_softmax_kernel<<<dim3(E, B), 256, 0, stream>>>(vu, iv, edge_adj, yita, alpha);

    // 3) aggregation GEMM + LeakyReLU (WMMA f32)
    agg_kernel<<<B * (E / 16), 32, 0, stream>>>(alpha, edge_attr, out);
}